// CLAHE_20151986553106
// MI455X (gfx1250) — compile-verified
//
#include <hip/hip_runtime.h>
#include <stdint.h>

// ---------------------------------------------------------------------------
// CLAHE for (32,1,1024,1024) f32, TILES 8x8 -> 2048 tiles of 128x128,
// NUM_BINS=256, CLIP_LIMIT=2.0 -> clip_count = 16384*2/256 = 128 exactly.
// One 256-thread workgroup per tile. Tile staged in LDS via TDM
// (tensor_load_to_lds), histogram via LDS atomics, 256-bin cumsum via chained
// V_WMMA_F32_16X16X4_F32 (scan as triangular-ones matmul), remap + coalesced
// float4 stores. HBM traffic ~256 MB total -> ~11 us roofline at 23.3 TB/s.
// ---------------------------------------------------------------------------

typedef float v2f __attribute__((ext_vector_type(2)));
typedef float v8f __attribute__((ext_vector_type(8)));
typedef unsigned int u32x4 __attribute__((ext_vector_type(4)));
typedef int i32x8 __attribute__((ext_vector_type(8)));
typedef int i32x4 __attribute__((ext_vector_type(4)));

#if defined(__has_builtin)
#  if __has_builtin(__builtin_amdgcn_tensor_load_to_lds) && \
      __has_builtin(__builtin_amdgcn_s_wait_tensorcnt)
#    define HAVE_TDM 1
#  else
#    define HAVE_TDM 0
#  endif
#else
#  define HAVE_TDM 0
#endif

__device__ __forceinline__ int clamp255(int i) {
  return i < 0 ? 0 : (i > 255 ? 255 : i);
}

__global__ __launch_bounds__(256) void clahe_tile_kernel(
    const float* __restrict__ img, float* __restrict__ out) {
  constexpr int TH = 128, TW = 128, NB = 256, W = 1024, H = 1024;
  constexpr float CLIPC = 128.0f;  // 16384 * 2.0 / 256

  __shared__ float    s_tile[TH * TW];   // 64 KB staged tile
  __shared__ unsigned s_hist[NB];
  __shared__ float    s_hc[NB];
  __shared__ float    s_cdf[NB];
  __shared__ float    s_T[16];
  __shared__ float    s_O[16];
  __shared__ unsigned s_excess;
  __shared__ float    s_inv;

  const int t   = blockIdx.x;
  const int b   = t >> 6;
  const int ty  = (t >> 3) & 7;
  const int tx  = t & 7;
  const int tid = threadIdx.x;

  const size_t tile_off = (size_t)b * (size_t)(H * W) +
                          (size_t)(ty * TH) * (size_t)W + (size_t)(tx * TW);
  const float* gsrc = img + tile_off;
  float*       gdst = out + tile_off;

  // ---------------- Phase 0: stage tile into LDS ----------------
#if HAVE_TDM
  if (tid < 32) {  // one TDM op issued by wave 0 (EXEC ignored by TDM)
    const uint64_t ga = (uint64_t)(uintptr_t)gsrc;
    u32x4 g0;
    g0[0] = 1u;                                   // count=1, user descriptor
    g0[1] = (uint32_t)(uintptr_t)(&s_tile[0]);    // LDS byte offset (low 32b of flat)
    g0[2] = (uint32_t)ga;                         // global_addr[31:0]
    g0[3] = ((uint32_t)(ga >> 32) & 0x01FFFFFFu)  // global_addr[56:32]
            | 0x80000000u;                        // type=2 ("image")
    i32x8 g1;
    g1[0] = (int)(2u << 16);          // workgroup_mask=0, data_size=2 (4B)
    g1[1] = (int)((unsigned)TW << 16);// tensor_dim0 = 128 (low16 at bits 63:48)
    g1[2] = (int)((unsigned)TH << 16);// tensor_dim0 hi=0 | tensor_dim1=128 low16
    g1[3] = (int)((unsigned)TW << 16);// tensor_dim1 hi=0 | tile_dim0=128
    g1[4] = TH;                       // tile_dim1=128 | tile_dim2=0
    g1[5] = W;                        // tensor_dim0_stride = 1024 elements
    g1[6] = 0;
    g1[7] = 0;
    i32x4 gz4 = {0, 0, 0, 0};
    i32x8 gz8 = {0, 0, 0, 0, 0, 0, 0, 0};
    __builtin_amdgcn_tensor_load_to_lds(g0, g1, gz4, gz4, gz8, 0);
    __builtin_amdgcn_s_wait_tensorcnt(0);
  }
#else
  {
    const float4* src4 = reinterpret_cast<const float4*>(gsrc);
    float4*       t4w  = reinterpret_cast<float4*>(s_tile);
    for (int k = tid; k < TH * TW / 4; k += 256) {
      const int r = k >> 5, c = k & 31;           // 32 float4 per tile row
      t4w[k] = src4[r * (W / 4) + c];
    }
  }
#endif
  s_hist[tid] = 0u;
  if (tid == 0) s_excess = 0u;
  __syncthreads();

  // ---------------- Phase 1: histogram (LDS atomics) ----------------
  const float4* t4 = reinterpret_cast<const float4*>(s_tile);
  for (int k = tid; k < TH * TW / 4; k += 256) {
    const float4 v = t4[k];
    atomicAdd(&s_hist[clamp255((int)(v.x * 256.0f))], 1u);
    atomicAdd(&s_hist[clamp255((int)(v.y * 256.0f))], 1u);
    atomicAdd(&s_hist[clamp255((int)(v.z * 256.0f))], 1u);
    atomicAdd(&s_hist[clamp255((int)(v.w * 256.0f))], 1u);
  }
  __syncthreads();

  // ---------------- Phase 2: clip + integer excess (deterministic) ----------
  {
    const unsigned h = s_hist[tid];
    if (h > 128u) atomicAdd(&s_excess, h - 128u);
  }
  __syncthreads();
  {
    const float redist = (float)s_excess * (1.0f / 256.0f);
    s_hc[tid] = fminf((float)s_hist[tid], CLIPC) + redist;
  }
  __syncthreads();

  // ---------------- Phase 3: 256-bin cumsum via WMMA f32 16x16x4 -----------
  // Hmat[r][c] = hc[16r+c].  P = Hmat x U (U upper-triangular ones) gives
  // intra-chunk inclusive prefixes; chunk offsets via 16-elem scalar scan.
  v8f acc = {0.f, 0.f, 0.f, 0.f, 0.f, 0.f, 0.f, 0.f};
  const int m  = tid & 15;
  const int hi = (tid >> 4) & 1;
  if (tid < 32) {
#pragma unroll
    for (int g = 0; g < 4; ++g) {
      const int k0 = 4 * g + 2 * hi;  // A/B 32-bit fragment layout per ISA
      v2f a, bm;
      a.x  = s_hc[16 * m + k0];
      a.y  = s_hc[16 * m + k0 + 1];
      bm.x = (k0     <= m) ? 1.0f : 0.0f;   // U[k0  ][n], n = lane&15
      bm.y = (k0 + 1 <= m) ? 1.0f : 0.0f;   // U[k0+1][n]
      acc = __builtin_amdgcn_wmma_f32_16x16x4_f32(
          false, a, false, bm, (short)0, acc, false, false);
    }
    if (m == 15) {  // lanes 15/31 hold chunk row-totals P[M][15]
#pragma unroll
      for (int v = 0; v < 8; ++v) s_T[8 * hi + v] = acc[v];
    }
  }
  __syncthreads();
  if (tid == 0) {
    float run = 0.0f;
#pragma unroll
    for (int i = 0; i < 16; ++i) { s_O[i] = run; run += s_T[i]; }
    s_inv = 1.0f / run;  // normalize by cdf[-1], as reference does
  }
  __syncthreads();
  if (tid < 32) {
    const float inv = s_inv;
#pragma unroll
    for (int v = 0; v < 8; ++v) {
      const int M = 8 * hi + v;  // C/D layout: vgpr v -> row v + 8*(lane>=16)
      s_cdf[16 * M + m] = (acc[v] + s_O[M]) * inv;
    }
  }
  __syncthreads();

  // ---------------- Phase 4: remap + coalesced store ----------------
  float4* dst4 = reinterpret_cast<float4*>(gdst);
  for (int k = tid; k < TH * TW / 4; k += 256) {
    const int r = k >> 5, c = k & 31;
    const float4 v = t4[k];
    float4 o;
    o.x = s_cdf[clamp255((int)(v.x * 255.0f))];
    o.y = s_cdf[clamp255((int)(v.y * 255.0f))];
    o.z = s_cdf[clamp255((int)(v.z * 255.0f))];
    o.w = s_cdf[clamp255((int)(v.w * 255.0f))];
    dst4[r * (W / 4) + c] = o;
  }
}

extern "C" void kernel_launch(void* const* d_in, const int* in_sizes, int n_in,
                              void* d_out, int out_size, void* d_ws, size_t ws_size,
                              hipStream_t stream) {
  (void)n_in; (void)out_size; (void)d_ws; (void)ws_size;
  const float* img = (const float*)d_in[0];
  float* out = (float*)d_out;
  const int B = in_sizes[0] / (1024 * 1024);  // 32
  const int n_tiles = B * 64;                 // 2048
  clahe_tile_kernel<<<dim3(n_tiles), dim3(256), 0, stream>>>(img, out);
}